// StemInference_86895778333054
// MI455X (gfx1250) — compile-verified
//
#include <hip/hip_runtime.h>

// Problem constants (match the reference).
#define B_  32
#define H_  512
#define W_  768
#define HW_ (H_ * W_)
#define RADIUS_  15.0f
#define THRESH_  0.1f

// ---------------------------------------------------------------------------
// Kernel 1: zero the vote accumulator with b128 stores.
// out_size = B*H*W = 12,582,912 floats = 3,145,728 float4 = 12288 blocks * 256.
// ---------------------------------------------------------------------------
__global__ __launch_bounds__(256) void StemInference_zero_kernel(float4* __restrict__ out) {
    int i = blockIdx.x * 256 + threadIdx.x;
    out[i] = make_float4(0.0f, 0.0f, 0.0f, 0.0f);
}

// ---------------------------------------------------------------------------
// Kernel 2: voting.
//   grid  = (H, B), block = 192 threads (= W/4 float4 lanes = 6 wave32s)
//   blockIdx.x = row y, blockIdx.y = batch b, threadIdx.x = x4-group.
// No integer division anywhere: all addressing is FMA on constants.
// Each thread: three b128 loads (kp, offx, offy), three CDNA5
// global_prefetch_b8 for 8 rows ahead, v_rndne_f32 vote rounding
// (round-half-to-even == jnp.round), up to 4 non-returning
// global_atomic_add_f32 scatters. Vote targets are Gaussian-local
// (sigma = 15 px) so the atomics hit L2-resident lines.
// ---------------------------------------------------------------------------
__global__ __launch_bounds__(192) void StemInference_vote_kernel(
    const float* __restrict__ kp,    // (B,1,H,W)
    const float* __restrict__ off,   // (B,2,H,W)
    float*       __restrict__ out)   // (B,H,W)
{
    const int y  = blockIdx.x;            // 0..H-1
    const int b  = blockIdx.y;            // 0..B-1
    const int x4 = (int)threadIdx.x * 4;  // 0..764, step 4

    const float* kp_b = kp + (size_t)b * HW_;
    const float* ox_b = off + (size_t)b * 2 * HW_;   // channel 0
    const float* oy_b = ox_b + HW_;                  // channel 1
    const int base = y * W_ + x4;                    // 16B aligned

    // b128 streaming loads of all three input streams.
    float4 w  = *(const float4*)(kp_b + base);
    float4 ox = *(const float4*)(ox_b + base);
    float4 oy = *(const float4*)(oy_b + base);

    // CDNA5 global_prefetch_b8: pull lines 8 rows ahead into cache.
    // Speculative prefetch -> tail-of-buffer overshoot silently dropped.
    __builtin_prefetch(kp_b + base + 8 * W_, 0, 1);
    __builtin_prefetch(ox_b + base + 8 * W_, 0, 1);
    __builtin_prefetch(oy_b + base + 8 * W_, 0, 1);

    float wv[4]  = { w.x,  w.y,  w.z,  w.w  };
    float oxv[4] = { ox.x, ox.y, ox.z, ox.w };
    float oyv[4] = { oy.x, oy.y, oy.z, oy.w };

    float* out_b = out + (size_t)b * HW_;
    const float fy = (float)y;

#pragma unroll
    for (int j = 0; j < 4; ++j) {
        // round(x + 15*off) with round-to-nearest-even (v_rndne_f32),
        // exactly matching jnp.round semantics.
        float fvx = rintf(fmaf(RADIUS_, oxv[j], (float)(x4 + j)));
        float fvy = rintf(fmaf(RADIUS_, oyv[j], fy));
        // Bounds-check in float (safe for arbitrarily large offsets), then cvt.
        bool valid = (fvx >= 0.0f) & (fvx < (float)W_) &
                     (fvy >= 0.0f) & (fvy < (float)H_) &
                     (wv[j] > THRESH_);
        if (valid) {
            int vx = (int)fvx;
            int vy = (int)fvy;
            atomicAdd(out_b + vy * W_ + vx, wv[j]);  // global_atomic_add_f32 (no return)
        }
    }
}

// ---------------------------------------------------------------------------
// Launch: zero pass, then vote pass, both on `stream` (graph-capture safe).
// ---------------------------------------------------------------------------
extern "C" void kernel_launch(void* const* d_in, const int* in_sizes, int n_in,
                              void* d_out, int out_size, void* d_ws, size_t ws_size,
                              hipStream_t stream) {
    const float* kp  = (const float*)d_in[0];   // (B,1,H,W) float32
    const float* off = (const float*)d_in[1];   // (B,2,H,W) float32
    float* out = (float*)d_out;                 // (B,H,W)   float32

    (void)in_sizes; (void)n_in; (void)d_ws; (void)ws_size; (void)out_size;

    const int n4 = (B_ * HW_) / 4;              // 3,145,728 float4
    const int zero_blocks = n4 / 256;           // 12288
    StemInference_zero_kernel<<<zero_blocks, 256, 0, stream>>>((float4*)out);

    dim3 vote_grid(H_, B_);                     // 512 x 32 blocks
    StemInference_vote_kernel<<<vote_grid, 192, 0, stream>>>(kp, off, out);
}